// GNN_64518998720823
// MI455X (gfx1250) — compile-verified
//
#include <hip/hip_runtime.h>
#include <hip/hip_bf16.h>
#include <math.h>

typedef float v2f __attribute__((ext_vector_type(2)));
typedef float v8f __attribute__((ext_vector_type(8)));
typedef int   v4i __attribute__((ext_vector_type(4)));

#define D   256
#define D3  768
#define TSTEPS 3
#define APITCH 260   // LDS row pitch in floats; 260 % 64 == 4 -> conflict-free A frags

#if __has_builtin(__builtin_amdgcn_global_load_async_to_lds_b128) && \
    __has_builtin(__builtin_amdgcn_s_wait_asynccnt)
#define HAVE_ASYNC_LDS 1
typedef v4i __attribute__((address_space(1)))* g_v4i_p;   // global int4*
typedef v4i __attribute__((address_space(3)))* l_v4i_p;   // LDS int4*
#else
#define HAVE_ASYNC_LDS 0
#endif

__device__ __forceinline__ float fast_rcp(float x) { return __builtin_amdgcn_rcpf(x); }
__device__ __forceinline__ float sigf(float x) { return fast_rcp(1.0f + __expf(-x)); }
__device__ __forceinline__ float tanh_fast(float x) {
  // 1 - 2/(e^{2x}+1); exp->inf gives +1, exp->0 gives -1 (correct saturation)
  return 1.0f - 2.0f * fast_rcp(__expf(2.0f * x) + 1.0f);
}

__device__ __forceinline__ v8f wmma4(v2f a, v2f b, v8f c) {
  // D = A(16x4,f32) x B(4x16,f32) + C(16x16,f32) on the matrix pipe, exact f32
  return __builtin_amdgcn_wmma_f32_16x16x4_f32(false, a, false, b, (short)0, c, false, false);
}

// B fragment of a row-major weight: b.x = W[kk][col], b.y = W[kk+1][col]
__device__ __forceinline__ v2f loadB(const float* __restrict__ W, int kk, int col, int ldw) {
  v2f b;
  b.x = W[(size_t)kk * ldw + col];
  b.y = W[(size_t)(kk + 1) * ldw + col];
  return b;
}

// Stage a 16xD row-major tile from global into LDS (pitch APITCH) with 256 threads.
__device__ __forceinline__ void stage16xD(const float* __restrict__ g, float* l, int tid) {
  const int row = tid >> 4, seg = tid & 15;
  const float* gs = g + (size_t)row * D + seg * 16;
  float* ls = l + row * APITCH + seg * 16;
#if HAVE_ASYNC_LDS
#pragma unroll
  for (int j = 0; j < 4; j++)
    __builtin_amdgcn_global_load_async_to_lds_b128(
        (g_v4i_p)(gs + 4 * j), (l_v4i_p)(ls + 4 * j), 0, 0);
#else
#pragma unroll
  for (int j = 0; j < 4; j++)
    *(float4*)(ls + 4 * j) = *(const float4*)(gs + 4 * j);
#endif
}

__device__ __forceinline__ void stage_wait() {
#if HAVE_ASYNC_LDS
  __builtin_amdgcn_s_wait_asynccnt(0);
#endif
  __syncthreads();
}

// ---------------- LayerNorm over last dim (one row per block, 256 threads) --------------
__global__ void ln_kernel(const float* __restrict__ src, float* __restrict__ dst,
                          const float* __restrict__ g, const float* __restrict__ b) {
  __shared__ float ps[8], ps2[8], mv[2];
  const int row = blockIdx.x;
  const int t = threadIdx.x;
  float x = src[(size_t)row * D + t];
  float s = x, s2 = x * x;
#pragma unroll
  for (int off = 16; off > 0; off >>= 1) {
    s += __shfl_xor(s, off);
    s2 += __shfl_xor(s2, off);
  }
  if ((t & 31) == 0) { ps[t >> 5] = s; ps2[t >> 5] = s2; }
  __syncthreads();
  if (t == 0) {
    float S = 0.0f, S2 = 0.0f;
#pragma unroll
    for (int j = 0; j < 8; j++) { S += ps[j]; S2 += ps2[j]; }
    float m = S / (float)D;
    float v = S2 / (float)D - m * m;
    mv[0] = m;
    mv[1] = rsqrtf(v + 1e-5f);
  }
  __syncthreads();
  dst[(size_t)row * D + t] = (x - mv[0]) * mv[1] * g[t] + b[t];
}

// ---------------- zero scratch ----------------------------------------------------------
__global__ void zero_kernel(float* __restrict__ p, long n) {
  long i = (long)blockIdx.x * blockDim.x + threadIdx.x;
  if (i < n) p[i] = 0.0f;
}

// ---------------- edge scatter: one wave per fact, 8 floats per lane --------------------
__global__ void edge_kernel(const float* __restrict__ q,         // [B,D] (this step)
                            const float* __restrict__ rel,       // [F,D]
                            const int* __restrict__ batch_ids,
                            const int* __restrict__ head_idx,
                            const int* __restrict__ tail_idx,
                            const float* __restrict__ prior_src, // [N]
                            float* __restrict__ neighbor,        // [N,D] accum
                            float* __restrict__ inter,           // [N] accum
                            int F) {
  const int f = blockIdx.x * 8 + (threadIdx.x >> 5);
  if (f >= F) return;
  const int lane = threadIdx.x & 31;

  // prefetch the next fact row handled by this wave slot (global_prefetch_b8)
  if (f + 8 < F) __builtin_prefetch(rel + (size_t)(f + 8) * D + lane * 8, 0, 1);

  const float prior = prior_src[head_idx[f]];
  if (prior == 0.0f) return;   // exact sparsity: labels are 0/1 or sigmoid(-1e20)==0

  const int b = batch_ids[f];
  const int t = tail_idx[f];
  const float4* r4 = (const float4*)(rel + (size_t)f * D + lane * 8);
  const float4* q4 = (const float4*)(q + (size_t)b * D + lane * 8);
  float* np = neighbor + (size_t)t * D + lane * 8;
  float4 ra = r4[0], rb = r4[1];
  float4 qa = q4[0], qb = q4[1];
  float va[8] = {ra.x * qa.x, ra.y * qa.y, ra.z * qa.z, ra.w * qa.w,
                 rb.x * qb.x, rb.y * qb.y, rb.z * qb.z, rb.w * qb.w};
#pragma unroll
  for (int j = 0; j < 8; j++) {
    if (va[j] > 0.0f) unsafeAtomicAdd(np + j, va[j] * prior);  // relu already applied
  }
  if (lane == 0) unsafeAtomicAdd(inter + t, prior);
}

// ---------------- fused dual-GEMM GRU + LayerNorm + score (WMMA f32) --------------------
// One block = 16 rows. 8 waves; wave w owns d-slices [16w,16w+16) and [128+16w,+16)
// across update/reset/memory chunks, so gates combine without cross-wave traffic.
__global__ void __launch_bounds__(256)
gru_kernel(const float* __restrict__ last,    // [N,D]  (LayerNorm'd neighbor)
           float* __restrict__ hidden,        // [N,D]  in/out
           const float* __restrict__ w_ih,    // [D,3D]
           const float* __restrict__ w_hh,    // [D,3D]
           const float* __restrict__ b_hh,    // [3D]
           const float* __restrict__ ln_g, const float* __restrict__ ln_b,
           const float* __restrict__ w_score, const float* __restrict__ b_score,
           const float* __restrict__ inter,        // [N]
           const float* __restrict__ prev_label,   // [N]
           const float* __restrict__ entity_mask,  // [N]
           float* __restrict__ out_label)          // [N]
{
  __shared__ float lds_last[16 * APITCH];
  __shared__ float lds_hid[16 * APITCH];
  __shared__ float lds_h[16 * APITCH];
  __shared__ float red1[256];
  __shared__ float red2[256];
  __shared__ float rmean[16];
  __shared__ float ristd[16];

  const int tid = threadIdx.x;
  const int w = tid >> 5;
  const int lane = tid & 31;
  const int half = lane >> 4;     // K-group for A/B frags, M-group for C/D
  const int m = lane & 15;        // A row within tile / C col within tile
  const int rowbase = blockIdx.x * 16;

  const int d0 = 16 * w;
  const int d1 = 128 + 16 * w;

  // stage A tiles (shared by all 8 waves) into LDS via async copy
  stage16xD(last + (size_t)rowbase * D, lds_last, tid);
  stage16xD(hidden + (size_t)rowbase * D, lds_hid, tid);
  stage_wait();

  v8f au0 = {}, au1 = {}, ar0 = {}, ar1 = {};
  v8f amx0 = {}, amh0 = {}, amx1 = {}, amh1 = {};

#pragma unroll 1
  for (int k0 = 0; k0 < D; k0 += 4) {
    const int kk = k0 + half * 2;
    v2f aL = *(const v2f*)(lds_last + m * APITCH + kk);  // ds_load_b64, bank-clean
    v2f aH = *(const v2f*)(lds_hid + m * APITCH + kk);

    // update gate (fused gx+gh)
    au0 = wmma4(aL, loadB(w_ih, kk, d0 + m, D3), au0);
    au0 = wmma4(aH, loadB(w_hh, kk, d0 + m, D3), au0);
    au1 = wmma4(aL, loadB(w_ih, kk, d1 + m, D3), au1);
    au1 = wmma4(aH, loadB(w_hh, kk, d1 + m, D3), au1);
    // reset gate (fused)
    ar0 = wmma4(aL, loadB(w_ih, kk, 256 + d0 + m, D3), ar0);
    ar0 = wmma4(aH, loadB(w_hh, kk, 256 + d0 + m, D3), ar0);
    ar1 = wmma4(aL, loadB(w_ih, kk, 256 + d1 + m, D3), ar1);
    ar1 = wmma4(aH, loadB(w_hh, kk, 256 + d1 + m, D3), ar1);
    // memory gate (kept separate: tanh(gx2 + r*gh2))
    amx0 = wmma4(aL, loadB(w_ih, kk, 512 + d0 + m, D3), amx0);
    amh0 = wmma4(aH, loadB(w_hh, kk, 512 + d0 + m, D3), amh0);
    amx1 = wmma4(aL, loadB(w_ih, kk, 512 + d1 + m, D3), amx1);
    amh1 = wmma4(aH, loadB(w_hh, kk, 512 + d1 + m, D3), amh1);
  }

  // add b_hh (belongs to the gh terms)
  const float bu0 = b_hh[d0 + m], bu1 = b_hh[d1 + m];
  const float br0 = b_hh[256 + d0 + m], br1 = b_hh[256 + d1 + m];
  const float bm0 = b_hh[512 + d0 + m], bm1 = b_hh[512 + d1 + m];
#pragma unroll
  for (int r = 0; r < 8; r++) {
    au0[r] += bu0; au1[r] += bu1;
    ar0[r] += br0; ar1[r] += br1;
    amh0[r] += bm0; amh1[r] += bm1;
  }

  // GRU elementwise; stage pre-LN hidden in LDS
#pragma unroll
  for (int r = 0; r < 8; r++) {
    const int rowInTile = r + 8 * half;
    {
      const int d = d0 + m;
      float upd = sigf(au0[r]);
      float rst = sigf(ar0[r]);
      float memv = tanh_fast(amx0[r] + rst * amh0[r]);
      float hp = lds_hid[rowInTile * APITCH + d];
      lds_h[rowInTile * APITCH + d] = (1.0f - upd) * memv + upd * hp;
    }
    {
      const int d = d1 + m;
      float upd = sigf(au1[r]);
      float rst = sigf(ar1[r]);
      float memv = tanh_fast(amx1[r] + rst * amh1[r]);
      float hp = lds_hid[rowInTile * APITCH + d];
      lds_h[rowInTile * APITCH + d] = (1.0f - upd) * memv + upd * hp;
    }
  }
  __syncthreads();

  // per-row LayerNorm: 16 threads per row, 16 elems per thread
  {
    const int row = tid >> 4, seg = tid & 15;
    float s = 0.0f, s2 = 0.0f;
#pragma unroll
    for (int j = 0; j < 16; j++) {
      float v = lds_h[row * APITCH + seg * 16 + j];
      s += v; s2 += v * v;
    }
    red1[tid] = s; red2[tid] = s2;
  }
  __syncthreads();
  if (tid < 16) {
    float s = 0.0f, s2 = 0.0f;
    for (int j = 0; j < 16; j++) { s += red1[tid * 16 + j]; s2 += red2[tid * 16 + j]; }
    float mean = s / (float)D;
    float var = s2 / (float)D - mean * mean;
    rmean[tid] = mean;
    ristd[tid] = rsqrtf(var + 1e-5f);
  }
  __syncthreads();
  {
    const int row = tid >> 4, seg = tid & 15;
    const int grow = rowbase + row;
    const float mean = rmean[row], istd = ristd[row];
    float sdot = 0.0f;
#pragma unroll
    for (int j = 0; j < 16; j++) {
      const int d = seg * 16 + j;
      float hn = (lds_h[row * APITCH + d] - mean) * istd * ln_g[d] + ln_b[d];
      hidden[(size_t)grow * D + d] = hn;
      sdot += hn * w_score[d];
    }
    red1[tid] = sdot;
  }
  __syncthreads();
  if (tid < 16) {
    const int grow = rowbase + tid;
    float s = 0.0f;
    for (int j = 0; j < 16; j++) s += red1[tid * 16 + j];
    s += b_score[0];
    float maskv = ((inter[grow] + prev_label[grow]) > 1e-8f) ? entity_mask[grow] : 0.0f;
    float sc = maskv * s + (1.0f - maskv) * (-1e20f);
    out_label[grow] = sigf(sc);
  }
}

// ---------------- final FFN: hidden @ w_ffn + b_ffn (WMMA f32) --------------------------
__global__ void __launch_bounds__(256)
ffn_kernel(const float* __restrict__ hidden, const float* __restrict__ w_ffn,
           const float* __restrict__ b_ffn, float* __restrict__ out) {
  __shared__ float lds_a[16 * APITCH];
  const int tid = threadIdx.x;
  const int w = tid >> 5, lane = tid & 31, half = lane >> 4, m = lane & 15;
  const int rowbase = blockIdx.x * 16;
  const int c0 = 16 * w, c1 = 128 + 16 * w;

  stage16xD(hidden + (size_t)rowbase * D, lds_a, tid);
  stage_wait();

  v8f a0 = {}, a1 = {};
#pragma unroll 1
  for (int k0 = 0; k0 < D; k0 += 4) {
    const int kk = k0 + half * 2;
    v2f a = *(const v2f*)(lds_a + m * APITCH + kk);
    a0 = wmma4(a, loadB(w_ffn, kk, c0 + m, D), a0);
    a1 = wmma4(a, loadB(w_ffn, kk, c1 + m, D), a1);
  }
  const float bb0 = b_ffn[c0 + m], bb1 = b_ffn[c1 + m];
#pragma unroll
  for (int r = 0; r < 8; r++) {
    const int grow = rowbase + r + 8 * half;
    out[(size_t)grow * D + c0 + m] = a0[r] + bb0;
    out[(size_t)grow * D + c1 + m] = a1[r] + bb1;
  }
}

extern "C" void kernel_launch(void* const* d_in, const int* in_sizes, int n_in,
                              void* d_out, int out_size, void* d_ws, size_t ws_size,
                              hipStream_t stream) {
  const float* instructions = (const float*)d_in[0];
  const float* entity_emb   = (const float*)d_in[1];
  const float* fact_rel     = (const float*)d_in[2];
  const float* topic_label  = (const float*)d_in[3];
  const float* entity_mask  = (const float*)d_in[4];
  const float* ln_g   = (const float*)d_in[5];
  const float* ln_b   = (const float*)d_in[6];
  const float* w_ih   = (const float*)d_in[7];
  const float* w_hh   = (const float*)d_in[8];
  const float* b_hh   = (const float*)d_in[9];
  const float* w_score = (const float*)d_in[10];
  const float* b_score = (const float*)d_in[11];
  const float* w_ffn  = (const float*)d_in[12];
  const float* b_ffn  = (const float*)d_in[13];
  const int* batch_ids = (const int*)d_in[14];
  const int* head_idx  = (const int*)d_in[15];
  const int* tail_idx  = (const int*)d_in[16];

  const int N = in_sizes[3];                 // B*E
  const int F = in_sizes[14];
  const int B = in_sizes[0] / (TSTEPS * D);

  float* out = (float*)d_out;
  float* hidden   = (float*)d_ws;            // N*D
  float* neighbor = hidden + (size_t)N * D;  // N*D
  float* inter    = neighbor + (size_t)N * D;// N

  // hidden = LayerNorm(entity_emb)
  ln_kernel<<<N, 256, 0, stream>>>(entity_emb, hidden, ln_g, ln_b);

  for (int step = 0; step < TSTEPS; ++step) {
    const float* prev = (step == 0) ? topic_label : (out + (size_t)(step - 1) * N);
    const long zc = (long)N * D + N;  // neighbor + inter are contiguous
    zero_kernel<<<(int)((zc + 255) / 256), 256, 0, stream>>>(neighbor, zc);
    edge_kernel<<<(F + 7) / 8, 256, 0, stream>>>(
        instructions + (size_t)step * B * D, fact_rel, batch_ids, head_idx,
        tail_idx, prev, neighbor, inter, F);
    ln_kernel<<<N, 256, 0, stream>>>(neighbor, neighbor, ln_g, ln_b);  // last
    gru_kernel<<<N / 16, 256, 0, stream>>>(
        neighbor, hidden, w_ih, w_hh, b_hh, ln_g, ln_b, w_score, b_score,
        inter, prev, entity_mask, out + (size_t)step * N);
  }

  ffn_kernel<<<N / 16, 256, 0, stream>>>(hidden, w_ffn, b_ffn,
                                         out + (size_t)TSTEPS * N);
}